// InfoLoss_91336774517494
// MI455X (gfx1250) — compile-verified
//
#include <hip/hip_runtime.h>
#include <hip/hip_bf16.h>

// ---------------------------------------------------------------------------
// InfoLoss on MI455X (gfx1250): chain-CMMD via Newton-Schulz inversion with
// native fp32 WMMA (V_WMMA_F32_16X16X4_F32). Heavy GEMMs use the CDNA5 async
// global->LDS path (GLOBAL_LOAD_ASYNC_TO_LDS_B128 + ASYNCcnt) with LDS double
// buffering: one barrier per K-stage, copies overlap WMMA compute.
// ---------------------------------------------------------------------------

#define NMAT 4096
#define NS_ITERS 30   // even -> result lands back in primary buffer

typedef float v2f __attribute__((ext_vector_type(2)));
typedef float v8f __attribute__((ext_vector_type(8)));

// The async-load builtin takes pointers to 16-byte int vectors with explicit
// address spaces: addrspace(1) = global source, addrspace(3) = LDS dest.
typedef int v4i_vs __attribute__((vector_size(4 * sizeof(int))));
typedef v4i_vs __attribute__((address_space(1)))* gvec_p;
typedef v4i_vs __attribute__((address_space(3)))* lvec_p;

__device__ __forceinline__ v8f wmma_f32_k4(v2f a, v2f b, v8f c) {
  // D = A(16x4,f32) * B(4x16,f32) + C(16x16,f32)
  return __builtin_amdgcn_wmma_f32_16x16x4_f32(
      /*neg_a=*/false, a, /*neg_b=*/false, b,
      /*c_mod=*/(short)0, c, /*reuse_a=*/false, /*reuse_b=*/false);
}

// async copy of 16 bytes: global -> LDS, tracked by ASYNCcnt.
// flat shared address low 32 bits == LDS byte offset (aperture mapping).
__device__ __forceinline__ void async_copy_b128(const float* g, void* lds) {
  __builtin_amdgcn_global_load_async_to_lds_b128(
      (gvec_p)(unsigned long long)(const void*)g,
      (lvec_p)(unsigned int)(unsigned long long)lds,
      0, 0);
}

__device__ __forceinline__ void wait_async0() {
  asm volatile("s_wait_asynccnt 0" ::: "memory");
}

__device__ __forceinline__ float waveSum(float v) {
#pragma unroll
  for (int off = 16; off > 0; off >>= 1) v += __shfl_xor(v, off, 32);
  return v;
}

// ---------------------------------------------------------------------------
// C = alpha * A@B (+ beta * Cin).  A,B,Cin,C all NMAT x NMAT row-major f32.
// 256 threads = 8 waves; 128x128 tile; wave = 64x32 (8 x 16x16 accumulators).
// Double-buffered LDS, async global->LDS copies, one barrier per stage.
// ---------------------------------------------------------------------------
__global__ __launch_bounds__(256) void k_gemm(
    const float* __restrict__ A, const float* __restrict__ B,
    const float* __restrict__ Cin, float* __restrict__ Cout,
    float alpha, float beta) {
  __shared__ __align__(16) float As[2][128][20];   // rows 80B -> B128 dst aligned
  __shared__ __align__(16) float Bs[2][16][132];   // rows 528B (16B multiple)
  const int t = threadIdx.x;
  const int lane = t & 31;
  const int w = t >> 5;
  const int bm = blockIdx.y << 7;
  const int bn = blockIdx.x << 7;
  const int wm = (w & 1) << 6;
  const int wn = (w >> 1) << 5;
  const int hl = lane & 15;
  const int hi = lane >> 4;

  v8f acc[4][2];
#pragma unroll
  for (int mi = 0; mi < 4; ++mi)
#pragma unroll
    for (int ni = 0; ni < 2; ++ni) acc[mi][ni] = {};

  // issue async copies for one 16-wide K stage into LDS buffer `buf`
  auto issueStage = [&](int buf, int k0) {
#pragma unroll
    for (int i = 0; i < 2; ++i) {                  // A tile: 512 B128, 2/thread
      const int idx = t * 2 + i;
      const int ar = idx >> 2;
      const int ac = (idx & 3) << 2;
      async_copy_b128(A + (size_t)(bm + ar) * NMAT + (k0 + ac), &As[buf][ar][ac]);
      const int br = idx >> 5;                     // B tile: 512 B128, 2/thread
      const int bc = (idx & 31) << 2;
      async_copy_b128(B + (size_t)(k0 + br) * NMAT + (bn + bc), &Bs[buf][br][bc]);
    }
  };

  issueStage(0, 0);
  for (int s = 0; s < NMAT / 16; ++s) {
    const int buf = s & 1;
    wait_async0();        // my copies into `buf` (and earlier) are done
    __syncthreads();      // everyone's copies done + prev reads of buf^1 done
    if (s + 1 < NMAT / 16) issueStage(buf ^ 1, (s + 1) * 16);

#pragma unroll
    for (int kk = 0; kk < 16; kk += 4) {
      const int ka = kk + hi * 2;                  // lanes 16-31 hold K=2,3
      v2f af[4], bf[2];
#pragma unroll
      for (int mi = 0; mi < 4; ++mi) {
        const int r = wm + mi * 16 + hl;
        af[mi].x = As[buf][r][ka]; af[mi].y = As[buf][r][ka + 1];
      }
#pragma unroll
      for (int ni = 0; ni < 2; ++ni) {
        const int c = wn + ni * 16 + hl;
        bf[ni].x = Bs[buf][ka][c]; bf[ni].y = Bs[buf][ka + 1][c];
      }
#pragma unroll
      for (int mi = 0; mi < 4; ++mi)
#pragma unroll
        for (int ni = 0; ni < 2; ++ni)
          acc[mi][ni] = wmma_f32_k4(af[mi], bf[ni], acc[mi][ni]);
    }
    // no trailing barrier: next stage's wait+barrier protects buffer reuse
  }

#pragma unroll
  for (int mi = 0; mi < 4; ++mi)
#pragma unroll
    for (int ni = 0; ni < 2; ++ni)
#pragma unroll
      for (int r = 0; r < 8; ++r) {                // C layout: VGPR r -> M=r / r+8
        const int row = bm + wm + mi * 16 + r + hi * 8;
        const int col = bn + wn + ni * 16 + hl;
        const size_t o = (size_t)row * NMAT + col;
        float v = alpha * acc[mi][ni][r];
        if (beta != 0.0f) v += beta * Cin[o];
        Cout[o] = v;
      }
}

// ---------------------------------------------------------------------------
// Out[i][j] = exp(-(nx[i]+ny[j]-2*x_i.y_j)*invDiv) (+ ridge on diagonal).
// K = 32 (one LDS stage). X rows at stride xs, Y rows at stride ys.
// ---------------------------------------------------------------------------
__global__ __launch_bounds__(256) void k_gram_rbf(
    const float* __restrict__ X, int xs, const float* __restrict__ Y, int ys,
    const float* __restrict__ nx, const float* __restrict__ ny,
    float* __restrict__ Out, float invDiv, float ridge) {
  __shared__ __align__(16) float Xs[128][34];
  __shared__ __align__(16) float Ys[128][34];
  const int t = threadIdx.x;
  const int lane = t & 31;
  const int w = t >> 5;
  const int bm = blockIdx.y << 7;
  const int bn = blockIdx.x << 7;
  const int wm = (w & 1) << 6;
  const int wn = (w >> 1) << 5;
  const int hl = lane & 15;
  const int hi = lane >> 4;

#pragma unroll
  for (int i = 0; i < 4; ++i) {               // 128x32 tiles: 1024 float4, 4/thread
    const int idx = t * 4 + i;
    const int row = idx >> 3;
    const int c4 = (idx & 7) << 2;
    const float4 v = *(const float4*)(X + (size_t)(bm + row) * xs + c4);
    Xs[row][c4 + 0] = v.x; Xs[row][c4 + 1] = v.y;
    Xs[row][c4 + 2] = v.z; Xs[row][c4 + 3] = v.w;
    const float4 u = *(const float4*)(Y + (size_t)(bn + row) * ys + c4);
    Ys[row][c4 + 0] = u.x; Ys[row][c4 + 1] = u.y;
    Ys[row][c4 + 2] = u.z; Ys[row][c4 + 3] = u.w;
  }
  __syncthreads();

  v8f acc[4][2];
#pragma unroll
  for (int mi = 0; mi < 4; ++mi)
#pragma unroll
    for (int ni = 0; ni < 2; ++ni) acc[mi][ni] = {};

#pragma unroll
  for (int kk = 0; kk < 32; kk += 4) {
    const int ka = kk + hi * 2;
    v2f af[4], bf[2];
#pragma unroll
    for (int mi = 0; mi < 4; ++mi) {
      const int r = wm + mi * 16 + hl;
      af[mi].x = Xs[r][ka]; af[mi].y = Xs[r][ka + 1];
    }
#pragma unroll
    for (int ni = 0; ni < 2; ++ni) {          // operand is Y^T: read Y rows
      const int c = wn + ni * 16 + hl;
      bf[ni].x = Ys[c][ka]; bf[ni].y = Ys[c][ka + 1];
    }
#pragma unroll
    for (int mi = 0; mi < 4; ++mi)
#pragma unroll
      for (int ni = 0; ni < 2; ++ni)
        acc[mi][ni] = wmma_f32_k4(af[mi], bf[ni], acc[mi][ni]);
  }

#pragma unroll
  for (int mi = 0; mi < 4; ++mi)
#pragma unroll
    for (int ni = 0; ni < 2; ++ni)
#pragma unroll
      for (int r = 0; r < 8; ++r) {
        const int row = bm + wm + mi * 16 + r + hi * 8;
        const int col = bn + wn + ni * 16 + hl;
        const float g = acc[mi][ni][r];
        float val = __expf(-(nx[row] + ny[col] - 2.0f * g) * invDiv);
        if (ridge != 0.0f && row == col) val += ridge;
        Out[(size_t)row * NMAT + col] = val;
      }
}

// ---------------------------------------------------------------------------
// Fused flat-MMD term: atomicAdd(acc, wgt * sum_ij exp(-(nx_i+ny_j-2 x_i.y_j)*invDiv))
// X,Y: (NMAT,256) row-major. Gram via WMMA, exp+reduce in registers.
// ---------------------------------------------------------------------------
__global__ __launch_bounds__(256) void k_mmd_flat(
    const float* __restrict__ X, const float* __restrict__ Y,
    const float* __restrict__ nx, const float* __restrict__ ny,
    float invDiv, float wgt, float* __restrict__ acc) {
  __shared__ __align__(16) float Xs[128][18];
  __shared__ __align__(16) float Ys[128][18];
  __shared__ float red[8];
  const int t = threadIdx.x;
  const int lane = t & 31;
  const int w = t >> 5;
  const int bm = blockIdx.y << 7;
  const int bn = blockIdx.x << 7;
  const int wm = (w & 1) << 6;
  const int wn = (w >> 1) << 5;
  const int hl = lane & 15;
  const int hi = lane >> 4;

  v8f accm[4][2];
#pragma unroll
  for (int mi = 0; mi < 4; ++mi)
#pragma unroll
    for (int ni = 0; ni < 2; ++ni) accm[mi][ni] = {};

  for (int k0 = 0; k0 < 256; k0 += 16) {
#pragma unroll
    for (int i = 0; i < 2; ++i) {
      const int idx = t * 2 + i;
      const int row = idx >> 2;
      const int c4 = (idx & 3) << 2;
      const float4 v = *(const float4*)(X + (size_t)(bm + row) * 256 + k0 + c4);
      Xs[row][c4 + 0] = v.x; Xs[row][c4 + 1] = v.y;
      Xs[row][c4 + 2] = v.z; Xs[row][c4 + 3] = v.w;
      const float4 u = *(const float4*)(Y + (size_t)(bn + row) * 256 + k0 + c4);
      Ys[row][c4 + 0] = u.x; Ys[row][c4 + 1] = u.y;
      Ys[row][c4 + 2] = u.z; Ys[row][c4 + 3] = u.w;
    }
    __syncthreads();
#pragma unroll
    for (int kk = 0; kk < 16; kk += 4) {
      const int ka = kk + hi * 2;
      v2f af[4], bf[2];
#pragma unroll
      for (int mi = 0; mi < 4; ++mi) {
        const int r = wm + mi * 16 + hl;
        af[mi].x = Xs[r][ka]; af[mi].y = Xs[r][ka + 1];
      }
#pragma unroll
      for (int ni = 0; ni < 2; ++ni) {
        const int c = wn + ni * 16 + hl;
        bf[ni].x = Ys[c][ka]; bf[ni].y = Ys[c][ka + 1];
      }
#pragma unroll
      for (int mi = 0; mi < 4; ++mi)
#pragma unroll
        for (int ni = 0; ni < 2; ++ni)
          accm[mi][ni] = wmma_f32_k4(af[mi], bf[ni], accm[mi][ni]);
    }
    __syncthreads();
  }

  float s = 0.0f;
#pragma unroll
  for (int mi = 0; mi < 4; ++mi)
#pragma unroll
    for (int ni = 0; ni < 2; ++ni)
#pragma unroll
      for (int r = 0; r < 8; ++r) {
        const int row = bm + wm + mi * 16 + r + hi * 8;
        const int col = bn + wn + ni * 16 + hl;
        const float g = accm[mi][ni][r];
        s += __expf(-(nx[row] + ny[col] - 2.0f * g) * invDiv);
      }
  s = waveSum(s);
  if (lane == 0) red[w] = s;
  __syncthreads();
  if (t == 0) {
    float tot = 0.0f;
#pragma unroll
    for (int k = 0; k < 8; ++k) tot += red[k];
    atomicAdd(acc, wgt * tot);
  }
}

// --------------------------- small helper kernels ---------------------------

__global__ void k_zero(float* __restrict__ p, int n) {
  const int i = blockIdx.x * blockDim.x + threadIdx.x;
  if (i < n) p[i] = 0.0f;
}

__global__ void k_build_pf(const float* __restrict__ sn, float* __restrict__ pf) {
  const int i = blockIdx.x * 256 + threadIdx.x;   // over N*V = 32768
  if (i >= NMAT * 8) return;
  const float v = sn[i];
  float4 q; q.x = v; q.y = v; q.z = v; q.w = v;
  float4* o = (float4*)(pf + (size_t)i * 32);
#pragma unroll
  for (int j = 0; j < 8; ++j) o[j] = q;
}

__global__ void k_norms(const float* __restrict__ X, int stride, int d,
                        float* __restrict__ out) {
  const int i = blockIdx.x * 256 + threadIdx.x;
  if (i >= NMAT) return;
  const float* r = X + (size_t)i * stride;
  float s = 0.0f;
  for (int j = 0; j < d; j += 4) {
    const float4 v = *(const float4*)(r + j);
    s += v.x * v.x + v.y * v.y + v.z * v.z + v.w * v.w;
  }
  out[i] = s;
}

__global__ __launch_bounds__(256) void k_dotsum(
    const float* __restrict__ A, const float* __restrict__ B,
    float wgt, float* __restrict__ acc) {
  const size_t NN = (size_t)NMAT * NMAT;
  const size_t step = (size_t)gridDim.x * 1024;
  float s = 0.0f;
  for (size_t i = ((size_t)blockIdx.x * 256 + threadIdx.x) * 4; i < NN; i += step) {
    const float4 a = *(const float4*)(A + i);
    const float4 b = *(const float4*)(B + i);
    s += a.x * b.x + a.y * b.y + a.z * b.z + a.w * b.w;
  }
  s = waveSum(s);
  __shared__ float red[8];
  if ((threadIdx.x & 31) == 0) red[threadIdx.x >> 5] = s;
  __syncthreads();
  if (threadIdx.x == 0) {
    float tot = 0.0f;
#pragma unroll
    for (int k = 0; k < 8; ++k) tot += red[k];
    atomicAdd(acc, wgt * tot);
  }
}

// max over rows of row-sum (entries positive -> ||K||_1 = ||K||_inf for SPD K)
__global__ __launch_bounds__(256) void k_rowmax(const float* __restrict__ A,
                                                float* __restrict__ acc) {
  const int row = blockIdx.x;
  float s = 0.0f;
  for (int c = threadIdx.x * 4; c < NMAT; c += 1024) {
    const float4 v = *(const float4*)(A + (size_t)row * NMAT + c);
    s += v.x + v.y + v.z + v.w;
  }
  s = waveSum(s);
  __shared__ float red[8];
  if ((threadIdx.x & 31) == 0) red[threadIdx.x >> 5] = s;
  __syncthreads();
  if (threadIdx.x == 0) {
    float tot = 0.0f;
#pragma unroll
    for (int k = 0; k < 8; ++k) tot += red[k];
    atomicMax((unsigned int*)acc, __float_as_uint(tot));  // positive floats
  }
}

__global__ void k_scale(const float* __restrict__ K, const float* __restrict__ s,
                        float* __restrict__ X) {
  const float nrm = s[0];
  const float inv = 1.0f / (nrm * nrm);   // X0 = K / (||K||_1 ||K||_inf)
  const size_t NN = (size_t)NMAT * NMAT;
  const size_t step = (size_t)gridDim.x * 1024;
  for (size_t i = ((size_t)blockIdx.x * 256 + threadIdx.x) * 4; i < NN; i += step) {
    const float4 v = *(const float4*)(K + i);
    float4 o; o.x = v.x * inv; o.y = v.y * inv; o.z = v.z * inv; o.w = v.w * inv;
    *(float4*)(X + i) = o;
  }
}

__global__ __launch_bounds__(256) void k_sqdiff(
    const float* __restrict__ a, const float* __restrict__ b, int n,
    float wgt, float* __restrict__ acc) {
  const size_t step = (size_t)gridDim.x * 1024;
  float s = 0.0f;
  for (size_t i = ((size_t)blockIdx.x * 256 + threadIdx.x) * 4; i < (size_t)n; i += step) {
    const float4 av = *(const float4*)(a + i);
    const float4 bv = *(const float4*)(b + i);
    const float dx = av.x - bv.x, dy = av.y - bv.y, dz = av.z - bv.z, dw = av.w - bv.w;
    s += dx * dx + dy * dy + dz * dz + dw * dw;
  }
  s = waveSum(s);
  __shared__ float red[8];
  if ((threadIdx.x & 31) == 0) red[threadIdx.x >> 5] = s;
  __syncthreads();
  if (threadIdx.x == 0) {
    float tot = 0.0f;
#pragma unroll
    for (int k = 0; k < 8; ++k) tot += red[k];
    atomicAdd(acc, wgt * tot);
  }
}

__global__ void k_finalize(const float* __restrict__ accs, float* __restrict__ out) {
  if (threadIdx.x == 0) { out[0] = accs[0]; out[1] = accs[1]; }
}

// ---------------------------------------------------------------------------
// Host orchestration. Workspace: 6 N^2 f32 buffers + pf + norm vectors + accs
// (~407 MB). All launches on `stream`, graph-capture safe.
// ---------------------------------------------------------------------------
extern "C" void kernel_launch(void* const* d_in, const int* in_sizes, int n_in,
                              void* d_out, int out_size, void* d_ws, size_t ws_size,
                              hipStream_t stream) {
  (void)in_sizes; (void)n_in; (void)out_size; (void)ws_size;
  const float* x  = (const float*)d_in[0];   // (N,256)
  const float* z  = (const float*)d_in[1];   // (N,8,32)
  const float* zz = (const float*)d_in[2];   // (N,8,32)
  const float* xx = (const float*)d_in[3];   // (N,256)
  const float* sn = (const float*)d_in[4];   // (N,8,1)
  float* out = (float*)d_out;
  float* ws  = (float*)d_ws;

  const size_t NN = (size_t)NMAT * NMAT;
  float* B0 = ws + 0 * NN;                   // Kq
  float* B1 = ws + 1 * NN;                   // Kp
  float* Xq = ws + 2 * NN;
  float* Xp = ws + 3 * NN;
  float* T  = ws + 4 * NN;                   // scratch
  float* S  = ws + 5 * NN;                   // scratch / NS ping-pong
  float* pf   = ws + 6 * NN;                 // (N,256) broadcast std_norm
  float* nqf  = pf + (size_t)NMAT * 256;     // flat norms
  float* npf  = nqf + NMAT;
  float* nzv  = npf + NMAT;                  // per-view norms of z  (3*N)
  float* nzzv = nzv + 3 * (size_t)NMAT;      // per-view norms of zz (3*N)
  float* accs = nzzv + 3 * (size_t)NMAT;     // [0]=mmd  [1]=nll  [2]=norm scalar

  const dim3 gt(32, 32);
  const float N2 = (float)NMAT * (float)NMAT;
  const float BETA = 1.0f, GAMMA = 500.0f, REG = 0.2f;

  k_zero<<<1, 64, 0, stream>>>(accs, 4);

  // ---- NLL: sum((xx-x)^2)/(2N) ----
  k_sqdiff<<<1024, 256, 0, stream>>>(xx, x, NMAT * 256, 1.0f / (2.0f * NMAT), accs + 1);

  // ---- Term 1: plain MMD over 256-dim flattened features, divisor d*V^2=2048 ----
  k_build_pf<<<(NMAT * 8) / 256, 256, 0, stream>>>(sn, pf);
  k_norms<<<NMAT / 256, 256, 0, stream>>>(z, 256, 256, nqf);
  k_norms<<<NMAT / 256, 256, 0, stream>>>(pf, 256, 256, npf);
  k_mmd_flat<<<gt, 256, 0, stream>>>(z,  z,  nqf, nqf, 1.0f / 2048.0f,  BETA / N2, accs);
  k_mmd_flat<<<gt, 256, 0, stream>>>(pf, pf, npf, npf, 1.0f / 2048.0f,  BETA / N2, accs);
  k_mmd_flat<<<gt, 256, 0, stream>>>(z,  pf, nqf, npf, 1.0f / 2048.0f, -2.0f * BETA / N2, accs);

  // ---- per-view squared norms for views 0..2 ----
  for (int v = 0; v < 3; ++v) {
    k_norms<<<NMAT / 256, 256, 0, stream>>>(z  + v * 32, 256, 32, nzv  + (size_t)v * NMAT);
    k_norms<<<NMAT / 256, 256, 0, stream>>>(zz + v * 32, 256, 32, nzzv + (size_t)v * NMAT);
  }

  // Newton-Schulz SPD inverse: X0 = K/||K||^2 ; X <- 2X - X(KX). Even #iters.
  auto invert = [&](const float* K, float* X, float* Xalt) {
    k_zero<<<1, 64, 0, stream>>>(accs + 2, 1);
    k_rowmax<<<NMAT, 256, 0, stream>>>(K, accs + 2);
    k_scale<<<4096, 256, 0, stream>>>(K, accs + 2, X);
    float* xa = X; float* xb = Xalt;
    for (int it = 0; it < NS_ITERS; ++it) {
      k_gemm<<<gt, 256, 0, stream>>>(K, xa, nullptr, T, 1.0f, 0.0f);        // T = K X
      k_gemm<<<gt, 256, 0, stream>>>(xa, T, xa, xb, -1.0f, 2.0f);           // X' = 2X - X T
      float* tmp = xa; xa = xb; xb = tmp;
    }
  };

  // ---- chain terms: (e,u) in {(0,1),(1,2),(0,2)} ----
  const int PE[3] = {0, 1, 0};
  const int PU[3] = {1, 2, 2};
  for (int p = 0; p < 3; ++p) {
    const int e = PE[p], u = PU[p];
    const float* qe = zz + e * 32; const float* peb = z + e * 32;
    const float* qu = zz + u * 32; const float* pub = z + u * 32;
    const float* nqe = nzzv + (size_t)e * NMAT; const float* npe = nzv + (size_t)e * NMAT;
    const float* nqu = nzzv + (size_t)u * NMAT; const float* npu = nzv + (size_t)u * NMAT;

    // Kq, Kp with ridge; then invert (Xq = Kq^-1, Xp = Kp^-1)
    k_gram_rbf<<<gt, 256, 0, stream>>>(qu, 256, qu, 256, nqu, nqu, B0, 1.0f / 32.0f, REG);
    k_gram_rbf<<<gt, 256, 0, stream>>>(pub, 256, pub, 256, npu, npu, B1, 1.0f / 32.0f, REG);
    invert(B0, Xq, S);
    invert(B1, Xp, S);

    // trprod(Lqq,Cqq) = sum(Lqq.Xq) - REG*sum((Lqq@Xq).Xq)  (Cqq = Xq - REG*Xq^2)
    k_gram_rbf<<<gt, 256, 0, stream>>>(qe, 256, qe, 256, nqe, nqe, T, 1.0f / 32.0f, 0.0f);
    k_dotsum<<<4096, 256, 0, stream>>>(T, Xq,  GAMMA / N2, accs);
    k_gemm<<<gt, 256, 0, stream>>>(T, Xq, nullptr, S, 1.0f, 0.0f);
    k_dotsum<<<4096, 256, 0, stream>>>(S, Xq, -REG * GAMMA / N2, accs);

    // trprod(Lpp,Cpp)
    k_gram_rbf<<<gt, 256, 0, stream>>>(peb, 256, peb, 256, npe, npe, T, 1.0f / 32.0f, 0.0f);
    k_dotsum<<<4096, 256, 0, stream>>>(T, Xp,  GAMMA / N2, accs);
    k_gemm<<<gt, 256, 0, stream>>>(T, Xp, nullptr, S, 1.0f, 0.0f);
    k_dotsum<<<4096, 256, 0, stream>>>(S, Xp, -REG * GAMMA / N2, accs);

    // -2*trprod(Lqp,Cqp) = -2*sum(Lqp . (Xq @ Kqp0 @ Xp))
    k_gram_rbf<<<gt, 256, 0, stream>>>(qu, 256, pub, 256, nqu, npu, T, 1.0f / 32.0f, 0.0f); // Kqp0
    k_gemm<<<gt, 256, 0, stream>>>(Xq, T, nullptr, S, 1.0f, 0.0f);                           // A = Xq Kqp0
    k_gemm<<<gt, 256, 0, stream>>>(S, Xp, nullptr, T, 1.0f, 0.0f);                           // B = A Xp
    k_gram_rbf<<<gt, 256, 0, stream>>>(qe, 256, peb, 256, nqe, npe, S, 1.0f / 32.0f, 0.0f);  // Lqp
    k_dotsum<<<4096, 256, 0, stream>>>(S, T, -2.0f * GAMMA / N2, accs);
  }

  k_finalize<<<1, 32, 0, stream>>>(accs, out);
}